// Attn_VarLevel_62053687492948
// MI455X (gfx1250) — compile-verified
//
#include <hip/hip_runtime.h>
#include <math.h>

typedef float v2f __attribute__((ext_vector_type(2)));
typedef float v8f __attribute__((ext_vector_type(8)));

#define B_ 16
#define P_ 96
#define T_ 128
#define V_ 64
#define N_ 16
#define D_ 128

// ---------------------------------------------------------------------------
// Async global -> LDS copy (CDNA5 ASYNCcnt path, no VGPR round-trip).
// The low 32 bits of a generic pointer to a __shared__ object are the
// wave-relative LDS byte address (ISA 10.2: LDS aperture keeps offset in
// addr[31:0]); the instruction's VDST wants exactly that.
// ---------------------------------------------------------------------------
__device__ __forceinline__ void async_ld_b128(const float* lds_dst,
                                              const float* gsrc)
{
    uint32_t loff = (uint32_t)(uintptr_t)lds_dst;
    asm volatile("global_load_async_to_lds_b128 %0, %1, off"
                 :: "v"(loff), "v"(gsrc) : "memory");
}
__device__ __forceinline__ void wait_async()
{
    asm volatile("s_wait_asynccnt 0x0" ::: "memory");
}

// ---------------------------------------------------------------------------
// Y[M x 128] = X[M x 128] @ W^T + bias      (W is [128 x 128] row-major)
// Block: 256 threads = 8 waves, 128 rows x 128 cols per block.
// Wave wv owns column strip [16*wv, 16*wv+16); B-fragments hoisted to
// registers and reused across all 8 row tiles.  FP32 WMMA 16x16x4.
// ---------------------------------------------------------------------------
__global__ __launch_bounds__(256) void gemm_dd_bias(
    const float* __restrict__ X, const float* __restrict__ W,
    const float* __restrict__ bias, float* __restrict__ Y)
{
    __shared__ float xs[128 * D_];  // 64 KB : A tile (128 rows)
    __shared__ float wl[D_ * D_];   // 64 KB : full weight matrix

    const int  tid      = threadIdx.x;
    const long row_base = (long)blockIdx.x * 128;

    // async-stage A tile and W (16 + 16 b128 ops per thread, ASYNCcnt <= 32)
    const float* xsrc = X + row_base * D_;
    for (int i = tid; i < 128 * D_ / 4; i += 256) {
        async_ld_b128(&xs[4 * i], xsrc + 4 * i);
        async_ld_b128(&wl[4 * i], W + 4 * i);
    }
    wait_async();
    __syncthreads();

    const int wv   = tid >> 5;      // wave id 0..7 -> column tile
    const int lane = tid & 31;
    const int hl   = lane >> 4;     // half-wave selects K pair (0,1)/(2,3)
    const int l    = lane & 15;
    const int col  = wv * 16 + l;

    // hoist all 32 B fragments: bf[k] = W^T[4k+2h .. +1][col]
    v2f bf[32];
    const float* brow = &wl[col * D_];
#pragma unroll
    for (int k = 0; k < 32; ++k)
        bf[k] = *(const v2f*)&brow[4 * k + 2 * hl];

    const float bb   = bias[col];
    float*      ydst = Y + row_base * D_;

    for (int rt = 0; rt < 8; ++rt) {
        const float* arow = &xs[(rt * 16 + l) * D_];
        v8f c = {};
#pragma unroll
        for (int k = 0; k < 32; ++k) {
            v2f a = *(const v2f*)&arow[4 * k + 2 * hl];
            c = __builtin_amdgcn_wmma_f32_16x16x4_f32(false, a, false, bf[k],
                                                      (short)0, c, false, false);
        }
#pragma unroll
        for (int r = 0; r < 8; ++r)
            ydst[(rt * 16 + r + 8 * hl) * D_ + col] = c[r] + bb;
    }
}

// ---------------------------------------------------------------------------
// Fused q-projection + gathered attention for one (b, p) slab.
// Block owns k[b, 32+p] (64x128) entirely: stages it in LDS, then overwrites
// the same global rows with the attention output (in-place "res" build).
// ---------------------------------------------------------------------------
__global__ __launch_bounds__(256) void qproj_attn(
    const float* __restrict__ Qin,   // queries (B,P,V,D)
    const int*   __restrict__ ccc,   // var_ccc (B,V,N)
    const float* __restrict__ Wq, const float* __restrict__ bq,
    float* __restrict__ Kbuf)        // k workspace (B,T,V,D)
{
    __shared__ float xs[V_ * D_];    // 32 KB queries slab
    __shared__ float qs[V_ * D_];    // 32 KB projected q
    __shared__ float ks[V_ * D_];    // 32 KB k slab
    __shared__ float attn[V_][N_];   //  4 KB softmax weights
    __shared__ int   jidx[V_][N_];   //  4 KB gathered indices

    const int tid = threadIdx.x;
    const int b   = blockIdx.x / P_;
    const int p   = blockIdx.x % P_;

    const float* qin  = Qin  + ((long)(b * P_ + p) * V_) * D_;
    float*       krow = Kbuf + ((long)(b * T_ + (T_ - P_) + p) * V_) * D_;

    for (int i = tid; i < V_ * D_ / 4; i += 256) {
        async_ld_b128(&xs[4 * i], qin + 4 * i);
        async_ld_b128(&ks[4 * i], krow + 4 * i);
    }
    wait_async();
    __syncthreads();

    // ---- q = xs @ Wq^T + bq  (64x128) via FP32 WMMA ----
    const int wv = tid >> 5, lane = tid & 31, hl = lane >> 4, l = lane & 15;
    const int col = wv * 16 + l;
    const float* brow = Wq + col * D_;
    const float  bqv  = bq[col];

    v2f bf[32];
#pragma unroll
    for (int k = 0; k < 32; ++k)
        bf[k] = *(const v2f*)&brow[4 * k + 2 * hl];

    for (int rt = 0; rt < 4; ++rt) {
        const float* arow = &xs[(rt * 16 + l) * D_];
        v8f c = {};
#pragma unroll
        for (int k = 0; k < 32; ++k) {
            v2f a = *(const v2f*)&arow[4 * k + 2 * hl];
            c = __builtin_amdgcn_wmma_f32_16x16x4_f32(false, a, false, bf[k],
                                                      (short)0, c, false, false);
        }
#pragma unroll
        for (int r = 0; r < 8; ++r)
            qs[(rt * 16 + r + 8 * hl) * D_ + col] = c[r] + bqv;
    }
    __syncthreads();

    // ---- scores + softmax: one half-wave (16 lanes) per v, lane = n ----
    const int  grp = tid >> 4;            // 0..15
    const int  nn  = tid & 15;            // n index
    const int* cb  = ccc + (b * V_) * N_;
    const float scale = 0.08838834764831845f;   // 1/sqrt(128)
    for (int v = grp; v < V_; v += 16) {
        const int j = cb[v * N_ + nn];
        const float* qv = &qs[v * D_];
        const float* kj = &ks[j * D_];
        float s = 0.f;
        for (int d = 0; d < D_; d += 4)
            s += qv[d] * kj[d] + qv[d + 1] * kj[d + 1]
               + qv[d + 2] * kj[d + 2] + qv[d + 3] * kj[d + 3];
        s *= scale;
        float m = s;
#pragma unroll
        for (int o = 1; o < 16; o <<= 1) m = fmaxf(m, __shfl_xor(m, o, 32));
        float e = __expf(s - m);
        float sum = e;
#pragma unroll
        for (int o = 1; o < 16; o <<= 1) sum += __shfl_xor(sum, o, 32);
        attn[v][nn] = e / sum;
        jidx[v][nn] = j;
    }
    __syncthreads();

    // ---- out[v][d] = sum_n attn * k_gather; overwrite k rows in place ----
    for (int v = wv * 8; v < wv * 8 + 8; ++v) {
#pragma unroll
        for (int dc = 0; dc < 4; ++dc) {
            const int d = dc * 32 + lane;
            float acc = 0.f;
#pragma unroll
            for (int n = 0; n < N_; ++n)
                acc += attn[v][n] * ks[jidx[v][n] * D_ + d];
            krow[v * D_ + d] = acc;
        }
    }
}

// ---------------------------------------------------------------------------
extern "C" void kernel_launch(void* const* d_in, const int* in_sizes, int n_in,
                              void* d_out, int out_size, void* d_ws, size_t ws_size,
                              hipStream_t stream)
{
    const float* queries = (const float*)d_in[0];
    const float* keys    = (const float*)d_in[1];
    const int*   ccc     = (const int*)d_in[2];
    const float* Wq      = (const float*)d_in[3];
    const float* bq      = (const float*)d_in[4];
    const float* Wkv     = (const float*)d_in[5];
    const float* bkv     = (const float*)d_in[6];
    const float* Wout    = (const float*)d_in[7];
    const float* bout    = (const float*)d_in[8];
    float* out  = (float*)d_out;
    float* kbuf = (float*)d_ws;               // B*T*V*D f32 = 64 MiB scratch

    (void)in_sizes; (void)n_in; (void)out_size; (void)ws_size;

    const int Mk = B_ * T_ * V_;              // 131072 rows

    // 1) k = keys @ Wkv^T + bkv    -> kbuf
    gemm_dd_bias<<<Mk / 128, 256, 0, stream>>>(keys, Wkv, bkv, kbuf);
    // 2) fused q-proj + gathered attention, writes res rows in place in kbuf
    qproj_attn<<<B_ * P_, 256, 0, stream>>>(queries, ccc, Wq, bq, kbuf);
    // 3) out = res @ Wout^T + bout
    gemm_dd_bias<<<Mk / 128, 256, 0, stream>>>(kbuf, Wout, bout, out);
}